// EEM_11948599017574
// MI455X (gfx1250) — compile-verified
//
#include <hip/hip_runtime.h>
#include <math.h>

// ---------------------------------------------------------------------------
// Fused channel-attention block for MI455X (gfx1250, wave32, WMMA).
//   K1: streaming conv stencils + 16x16xN Gram via V_WMMA_F32_16X16X4_F32
//   K2: deterministic reduce + softmax + algebraic fold into a 3x3 conv
//   K3: out = x + FW (*) x   (pure streaming, memory-bound)
// ---------------------------------------------------------------------------

#define HH 512
#define WW 512
#define BB 2
#define HEADS 8
#define HWN (HH * WW)

typedef float v2f __attribute__((ext_vector_type(2)));
typedef float v8f __attribute__((ext_vector_type(8)));

#define LROW 516  // 514 valid cols (-1..512), padded

// ---------------------------------------------------------------------------
// Kernel 1: per-(b,head,strip) Gram partials + squared norms.
// Block = 256 threads = 8 waves; wave w handles head w. Grid = (nstrips, B).
// LDS ring buffer of 3 image rows (x0, x1, e), zero-padded borders.
// ---------------------------------------------------------------------------
__global__ __launch_bounds__(256)
void eem_gram(const float* __restrict__ x, const float* __restrict__ e,
              const float* __restrict__ w_conv1, const float* __restrict__ w_dw,
              const float* __restrict__ w_ehead,
              float* __restrict__ Gp, float* __restrict__ SQp,
              float* __restrict__ SKp, int nstrips, int rowsPer) {
  __shared__ float sx0[3 * LROW];
  __shared__ float sx1[3 * LROW];
  __shared__ float se [3 * LROW];

  const int tid  = threadIdx.x;
  const int h    = tid >> 5;      // head = wave id
  const int lane = tid & 31;
  const int b    = blockIdx.y;
  const int s    = blockIdx.x;
  const int y0   = s * rowsPer;

  const float* gx0 = x + (size_t)(b * 2 + 0) * HWN;
  const float* gx1 = x + (size_t)(b * 2 + 1) * HWN;
  const float* ge  = e + (size_t)b * HWN;

  // Per-lane channel weights (channel = h*16 + (lane&15); both half-waves share).
  const int ch  = lane & 15;
  const int qch = h * 16 + ch;
  const float ca = w_conv1[qch * 2 + 0];
  const float cb = w_conv1[qch * 2 + 1];
  float wq0[9], wq1[9], we9[9];
#pragma unroll
  for (int t = 0; t < 9; ++t) {
    const float dwv = w_dw[qch * 9 + t];
    wq0[t] = dwv * ca;
    wq1[t] = dwv * cb;
    we9[t] = w_ehead[qch * 9 + t];
  }

  auto loadRow = [&](int yy, int slot) {
    for (int i = tid; i < 514; i += 256) {
      const int col = i - 1;
      float v0 = 0.f, v1 = 0.f, ve = 0.f;
      if (yy >= 0 && yy < HH && col >= 0 && col < WW) {
        const int g = yy * WW + col;
        v0 = gx0[g]; v1 = gx1[g]; ve = ge[g];
      }
      sx0[slot * LROW + i] = v0;
      sx1[slot * LROW + i] = v1;
      se [slot * LROW + i] = ve;
    }
  };

  // Preload rows y0-1, y0, y0+1 into their ring slots.
  loadRow(y0 - 1, (y0 + 2) % 3);
  loadRow(y0,     (y0    ) % 3);
  loadRow(y0 + 1, (y0 + 1) % 3);

  const int pair = lane >> 4;   // 0: pixel pair (0,1) for A / (0,2) for B; 1: other half
  v8f acc = {0.f, 0.f, 0.f, 0.f, 0.f, 0.f, 0.f, 0.f};
  float sq = 0.f, sk = 0.f;

  for (int y = y0; y < y0 + rowsPer; ++y) {
    __syncthreads();  // new row visible

    const float* r0[3]; const float* r1[3]; const float* re[3];
#pragma unroll
    for (int dy = 0; dy < 3; ++dy) {
      const int slot = (y + dy + 2) % 3;   // global row y+dy-1
      r0[dy] = &sx0[slot * LROW];
      r1[dy] = &sx1[slot * LROW];
      re[dy] = &se [slot * LROW];
    }

    for (int j = 0; j < WW / 4; ++j) {
      const int c0 = 4 * j;
      const int pA = c0 + 2 * pair;  // A: q at cols pA, pA+1  (K = 2*pair, 2*pair+1)
      const int pB = c0 + pair;      // B: k at cols pB, pB+2  (K = pair, pair+2)
      float ax = 0.f, ay = 0.f, bx = 0.f, by = 0.f;
#pragma unroll
      for (int dy = 0; dy < 3; ++dy) {
        const float* R0 = r0[dy]; const float* R1 = r1[dy]; const float* RE = re[dy];
#pragma unroll
        for (int dx = 0; dx < 3; ++dx) {
          const int t = dy * 3 + dx;
          ax = fmaf(wq0[t], R0[pA + dx],     fmaf(wq1[t], R1[pA + dx],     ax));
          ay = fmaf(wq0[t], R0[pA + 1 + dx], fmaf(wq1[t], R1[pA + 1 + dx], ay));
          bx = fmaf(we9[t], RE[pB + dx],     bx);
          by = fmaf(we9[t], RE[pB + 2 + dx], by);
        }
      }
      sq += ax * ax + ay * ay;
      sk += bx * bx + by * by;
      v2f A = {ax, ay};
      v2f Bm = {bx, by};
      // D = A(16x4) * B(4x16) + C   -- v_wmma_f32_16x16x4_f32
      acc = __builtin_amdgcn_wmma_f32_16x16x4_f32(false, A, false, Bm, (short)0,
                                                  acc, false, false);
    }

    __syncthreads();  // done reading slot (y-1)%3
    if (y < y0 + rowsPer - 1) loadRow(y + 2, (y + 2) % 3);
  }

  // Write deterministic partials: slot (b, strip, head).
  const size_t pb = ((size_t)b * nstrips + s) * HEADS + h;
  float* gdst = Gp + pb * 256;
#pragma unroll
  for (int v = 0; v < 8; ++v) {
    const int row = v + (pair ? 8 : 0);     // C layout: VGPR v -> M=v (lanes<16) / v+8
    gdst[row * 16 + ch] = acc[v];
  }
  const float sq2 = sq + __shfl_xor(sq, 16, 32);
  const float sk2 = sk + __shfl_xor(sk, 16, 32);
  if (lane < 16) {
    SQp[pb * 16 + ch] = sq2;
    SKp[pb * 16 + ch] = sk2;
  }
}

// ---------------------------------------------------------------------------
// Kernel 2: fixed-order reduction -> softmax(attn) -> fold proj+dw+conv1 into
// 72 effective 3x3 conv weights FW[b][o][i][t]. Single block, 256 threads.
// ---------------------------------------------------------------------------
__global__ __launch_bounds__(256)
void eem_attn_fold(const float* __restrict__ Gp, const float* __restrict__ SQp,
                   const float* __restrict__ SKp,
                   const float* __restrict__ w_proj, const float* __restrict__ w_dw,
                   const float* __restrict__ w_conv1, const float* __restrict__ a1,
                   float* __restrict__ FWg, int nstrips) {
  __shared__ float gA[16 * 256];   // [bh][c][d] : Gram, then attn in-place
  __shared__ float sqL[256];       // [bh][c]
  __shared__ float skL[256];       // [bh][d]
  __shared__ float ML[512];        // [b][o][128]

  const int tid = threadIdx.x;

  // Reduce Gram partials (fixed strip order => deterministic).
  for (int idx = tid; idx < 4096; idx += 256) {
    const int bh = idx >> 8;
    const int b = bh >> 3, h = bh & 7;
    float sum = 0.f;
    for (int st = 0; st < nstrips; ++st)
      sum += Gp[(((size_t)b * nstrips + st) * HEADS + h) * 256 + (idx & 255)];
    gA[idx] = sum;
  }
  {
    const int bh = tid >> 4, c = tid & 15;
    const int b = bh >> 3, h = bh & 7;
    float s1 = 0.f, s2 = 0.f;
    for (int st = 0; st < nstrips; ++st) {
      const size_t pb = ((size_t)b * nstrips + st) * HEADS + h;
      s1 += SQp[pb * 16 + c];
      s2 += SKp[pb * 16 + c];
    }
    sqL[tid] = s1;
    skL[tid] = s2;
  }
  __syncthreads();

  // Per-row softmax: thread = (bh, c). F.normalize eps = 1e-12 on each norm.
  {
    const int bh = tid >> 4, c = tid & 15;
    const int h = bh & 7;
    const float nq = fmaxf(sqrtf(sqL[bh * 16 + c]), 1e-12f);
    const float sc = a1[h];
    float row[16], mx = -3.4e38f;
#pragma unroll
    for (int d = 0; d < 16; ++d) {
      const float nk = fmaxf(sqrtf(skL[bh * 16 + d]), 1e-12f);
      row[d] = gA[bh * 256 + c * 16 + d] / (nq * nk) * sc;
      mx = fmaxf(mx, row[d]);
    }
    float den = 0.f;
#pragma unroll
    for (int d = 0; d < 16; ++d) { row[d] = expf(row[d] - mx); den += row[d]; }
    const float inv = 1.f / den;
#pragma unroll
    for (int d = 0; d < 16; ++d) gA[bh * 256 + c * 16 + d] = row[d] * inv;
  }
  __syncthreads();

  // M[b][o][h*16+d] = sum_c w_proj[o][h*16+c] * attn[b,h,c,d]
  for (int m = tid; m < 512; m += 256) {
    const int b = m >> 8, o = (m >> 7) & 1, jj = m & 127;
    const int h = jj >> 4, d = jj & 15;
    const int bh = b * 8 + h;
    float s = 0.f;
#pragma unroll
    for (int c = 0; c < 16; ++c)
      s += w_proj[o * 128 + h * 16 + c] * gA[bh * 256 + c * 16 + d];
    ML[m] = s;
  }
  __syncthreads();

  // FW[b][o][i][t] = sum_j M[b][o][j] * w_dw[128+j][t] * w_conv1[128+j][i]
  for (int f = tid; f < 72; f += 256) {
    const int b = f / 36;
    int r = f % 36;
    const int o = r / 18; r %= 18;
    const int i = r / 9;
    const int t = r % 9;
    float s = 0.f;
    for (int j = 0; j < 128; ++j)
      s += ML[((b * 2 + o) << 7) + j] * w_dw[(128 + j) * 9 + t] *
           w_conv1[(128 + j) * 2 + i];
    FWg[f] = s;
  }
}

// ---------------------------------------------------------------------------
// Kernel 3: out[b,o] = x[b,o] + sum_i FW[b,o,i] (*) x[b,i]   (3x3, SAME).
// Pure streaming; FW broadcast via LDS. Grid = (HW/256, B).
// ---------------------------------------------------------------------------
__global__ __launch_bounds__(256)
void eem_out(const float* __restrict__ x, const float* __restrict__ FWg,
             float* __restrict__ out) {
  __shared__ float fw[36];
  const int tid = threadIdx.x;
  const int b = blockIdx.y;
  if (tid < 36) fw[tid] = FWg[b * 36 + tid];
  __syncthreads();

  const int p = blockIdx.x * 256 + tid;
  const int y = p >> 9, xc = p & 511;
  const float* x0 = x + (size_t)(b * 2 + 0) * HWN;
  const float* x1 = x + (size_t)(b * 2 + 1) * HWN;

  // prefetch a couple of rows ahead (speculative; gfx1250 global_prefetch_b8)
  const int pp = (p + 2 * WW < HWN) ? (p + 2 * WW) : (HWN - 1);
  __builtin_prefetch(&x0[pp], 0, 1);
  __builtin_prefetch(&x1[pp], 0, 1);

  float a0 = x0[p];
  float a1v = x1[p];
#pragma unroll
  for (int dy = -1; dy <= 1; ++dy) {
    const int yy = y + dy;
    const bool ry = (unsigned)yy < (unsigned)HH;
#pragma unroll
    for (int dx = -1; dx <= 1; ++dx) {
      const int xx = xc + dx;
      const bool ok = ry && ((unsigned)xx < (unsigned)WW);
      const int t = (dy + 1) * 3 + (dx + 1);
      const int g = yy * WW + xx;
      const float v0 = ok ? x0[g] : 0.f;
      const float v1 = ok ? x1[g] : 0.f;
      a0  += fw[0 + t] * v0 + fw[9 + t] * v1;
      a1v += fw[18 + t] * v0 + fw[27 + t] * v1;
    }
  }
  out[(size_t)(b * 2 + 0) * HWN + p] = a0;
  out[(size_t)(b * 2 + 1) * HWN + p] = a1v;
}

// ---------------------------------------------------------------------------
extern "C" void kernel_launch(void* const* d_in, const int* in_sizes, int n_in,
                              void* d_out, int out_size, void* d_ws, size_t ws_size,
                              hipStream_t stream) {
  (void)in_sizes; (void)n_in; (void)out_size;
  const float* x       = (const float*)d_in[0];
  const float* e       = (const float*)d_in[1];
  const float* w_conv1 = (const float*)d_in[2];
  const float* w_dw    = (const float*)d_in[3];
  const float* w_ehead = (const float*)d_in[4];
  const float* w_proj  = (const float*)d_in[5];
  const float* a1      = (const float*)d_in[6];
  float* ws = (float*)d_ws;

  // Pick strip count (power of two) so deterministic partials fit in ws.
  int nstrips = 128;
  while (nstrips > 1) {
    const size_t needF = (size_t)BB * nstrips * HEADS * (256 + 32) + 72;
    if (needF * sizeof(float) <= ws_size) break;
    nstrips >>= 1;
  }
  const int rowsPer = HH / nstrips;

  float* Gp  = ws;
  float* SQp = Gp  + (size_t)BB * nstrips * HEADS * 256;
  float* SKp = SQp + (size_t)BB * nstrips * HEADS * 16;
  float* FWg = SKp + (size_t)BB * nstrips * HEADS * 16;

  eem_gram<<<dim3(nstrips, BB), 256, 0, stream>>>(
      x, e, w_conv1, w_dw, w_ehead, Gp, SQp, SKp, nstrips, rowsPer);

  eem_attn_fold<<<1, 256, 0, stream>>>(
      Gp, SQp, SKp, w_proj, w_dw, w_conv1, a1, FWg, nstrips);

  eem_out<<<dim3(HWN / 256, BB), 256, 0, stream>>>(x, FWg, (float*)d_out);
}